// Prob_Loss_20615843021101
// MI455X (gfx1250) — compile-verified
//
#include <hip/hip_runtime.h>
#include <hip/hip_bf16.h>
#include <math.h>

typedef float v2f __attribute__((ext_vector_type(2)));
typedef float v4f __attribute__((ext_vector_type(4)));
typedef float v8f __attribute__((ext_vector_type(8)));

#define N_ROWS 8192
#define D_COLS 4096

// Reduce two per-lane fp32 partials (sq, vr) across a wave32 with one
// V_WMMA_F32_16X16X4_F32.
//
// A-matrix 16x4 layout (ISA 7.12.2): vgpr0 lanes0-15 -> A[m][0], lanes16-31 ->
// A[m][2]; vgpr1 -> A[m][1]/A[m][3]. So a = {sq, vr} gives row m =
// {sq[m], vr[m], sq[m+16], vr[m+16]}.
// B-matrix 4x16 selector: B[0][n]=B[2][n]=(n<8), B[1][n]=B[3][n]=(n>=8):
//   D[m][n<8]  = sq[m] + sq[m+16]
//   D[m][n>=8] = vr[m] + vr[m+16]
// Sum D's 8 VGPRs per lane, fold lane^16, then lanes 0-7 hold total sq and
// lanes 8-15 hold total vr.
__device__ __forceinline__ void waveReducePairWmma(float sq, float vr,
                                                   float& sqTot, float& vrTot) {
  const int lane = threadIdx.x & 31;
  v2f a;
  a.x = sq;
  a.y = vr;
  v2f b;
  b.x = (lane & 8) ? 0.0f : 1.0f;
  b.y = (lane & 8) ? 1.0f : 0.0f;
  v8f c = {};
  v8f d = __builtin_amdgcn_wmma_f32_16x16x4_f32(
      /*neg_a=*/false, a, /*neg_b=*/false, b,
      /*c_mod=*/(short)0, c, /*reuse_a=*/false, /*reuse_b=*/false);
  float s = d[0] + d[1] + d[2] + d[3] + d[4] + d[5] + d[6] + d[7];
  s += __shfl_xor(s, 16, 32);
  sqTot = s;                 // valid in lanes 0..7
  vrTot = __shfl(s, 8, 32);  // lanes 8..15 all hold var total; broadcast
}

__global__ void __launch_bounds__(256) lossInitKernel(float* accum) {
  if (threadIdx.x < 2) accum[threadIdx.x] = 0.0f;
}

// Streams 384 MB once: mean/var from inputs (N x 2D), targets (N x D).
// Non-temporal b128 loads (no reuse; working set >> 192MB L2).
__global__ void __launch_bounds__(256)
lossMainKernel(const float* __restrict__ inputs,
               const float* __restrict__ targets,
               float* __restrict__ accum) {
  constexpr unsigned kDV = D_COLS / 4;                  // 1024 float4 per row
  constexpr unsigned kTotalV = (unsigned)N_ROWS * kDV;  // 8388608 float4s
  const unsigned tid = blockIdx.x * blockDim.x + threadIdx.x;
  const unsigned stride = gridDim.x * blockDim.x;

  float sq = 0.0f;  // sum of diff^2 / var
  float vs = 0.0f;  // sum of var

  for (unsigned v = tid; v < kTotalV; v += stride) {
    const unsigned row = v >> 10;                 // v / (D/4)
    const unsigned col = (v & (kDV - 1u)) << 2;   // element offset in row
    const size_t inBase = (size_t)row * (2u * D_COLS) + col;
    const size_t tgBase = (size_t)row * D_COLS + col;

    v4f m = __builtin_nontemporal_load((const v4f*)(inputs + inBase));
    v4f w = __builtin_nontemporal_load((const v4f*)(inputs + inBase + D_COLS));
    v4f t = __builtin_nontemporal_load((const v4f*)(targets + tgBase));

    v4f dlt = m - t;
    v4f r;
    r.x = __builtin_amdgcn_rcpf(w.x);
    r.y = __builtin_amdgcn_rcpf(w.y);
    r.z = __builtin_amdgcn_rcpf(w.z);
    r.w = __builtin_amdgcn_rcpf(w.w);
    v4f term = dlt * dlt * r;
    sq += term.x + term.y + term.z + term.w;
    vs += w.x + w.y + w.z + w.w;
  }

  // Wave32 reduction through the matrix pipe (EXEC all-1s here: no divergence).
  float sqTot, vrTot;
  waveReducePairWmma(sq, vs, sqTot, vrTot);

  __shared__ float ldsSq[8];
  __shared__ float ldsVr[8];
  const int wid = threadIdx.x >> 5;
  if ((threadIdx.x & 31) == 0) {
    ldsSq[wid] = sqTot;
    ldsVr[wid] = vrTot;
  }
  __syncthreads();
  if (threadIdx.x == 0) {
    float aSq = 0.0f, aVr = 0.0f;
#pragma unroll
    for (int i = 0; i < 8; ++i) {
      aSq += ldsSq[i];
      aVr += ldsVr[i];
    }
    atomicAdd(&accum[0], aSq);
    atomicAdd(&accum[1], aVr);
  }
}

__global__ void __launch_bounds__(32)
lossFinalizeKernel(const float* __restrict__ accum, float* __restrict__ out) {
  if (threadIdx.x == 0) {
    out[0] = accum[0] + logf(accum[1]);
  }
}

extern "C" void kernel_launch(void* const* d_in, const int* in_sizes, int n_in,
                              void* d_out, int out_size, void* d_ws,
                              size_t ws_size, hipStream_t stream) {
  const float* inputs = (const float*)d_in[0];   // (N, 2D) fp32
  const float* targets = (const float*)d_in[1];  // (N, D)  fp32
  float* accum = (float*)d_ws;                   // [0]=sum sq, [1]=sum var
  float* out = (float*)d_out;                    // scalar fp32

  lossInitKernel<<<1, 32, 0, stream>>>(accum);
  lossMainKernel<<<2048, 256, 0, stream>>>(inputs, targets, accum);
  lossFinalizeKernel<<<1, 32, 0, stream>>>(accum, out);
}